// TransducerPrediction_29128468202004
// MI455X (gfx1250) — compile-verified
//
#include <hip/hip_runtime.h>
#include <math.h>

typedef __attribute__((ext_vector_type(2))) float v2f;
typedef __attribute__((ext_vector_type(8))) float v8f;

#define B_   64
#define U_   512
#define V_   1024
#define D_   512
#define H_   640
#define FH_  2560   // 4*H
#define P_   512
#define M_   (B_ * U_)   // 32768
#define LN_EPS 0.001f

#define WROW_ 80     // padded LDS row stride (floats): rows kk,kk+2 -> +32 banks

__device__ __forceinline__ v8f wmma_f32_k4(v2f a, v2f b, v8f c) {
    // D = A(16x4,f32) * B(4x16,f32) + C(16x16,f32)
    return __builtin_amdgcn_wmma_f32_16x16x4_f32(
        /*neg_a=*/false, a, /*neg_b=*/false, b,
        /*c_mod=*/(short)0, c, /*reuse_a=*/false, /*reuse_b=*/false);
}

__device__ __forceinline__ void async_wait0() {
#if __has_builtin(__builtin_amdgcn_s_wait_asynccnt)
    __builtin_amdgcn_s_wait_asynccnt(0);
#else
    asm volatile("s_wait_asynccnt 0x0" ::: "memory");
#endif
}

// ---------------------------------------------------------------------------
// Generic GEMM:  Out(M,N) = A(M,K) @ W(K,N) + bias(N)
// If GATHER: A[m][k] = embed[tokens[m]*K + k]   (fused embedding lookup)
// One wave computes a 16x64 output strip (4 WMMA tiles along N, shared A frag).
// block = 256 threads (8 waves); grid = (M/128, N/64).
// All 8 waves share ncol0, so the W tile (16K x 64N) is staged once per block
// into LDS with async global->LDS b128 copies, double buffered: the fetch of
// tile kb+1 overlaps the 4 WMMAs per N-tile on tile kb.
// ---------------------------------------------------------------------------
template <bool GATHER>
__global__ void gemm_bias_kernel(const float* __restrict__ A,
                                 const int*   __restrict__ tokens,
                                 const float* __restrict__ embed,
                                 const float* __restrict__ W,
                                 const float* __restrict__ bias,
                                 float* __restrict__ Out,
                                 int M, int K, int N) {
    __shared__ float wbuf[2][16 * WROW_];

    const int tid   = threadIdx.x;
    const int lane  = tid & 31;
    const int wave  = tid >> 5;
    const int mtile = blockIdx.x * 8 + wave;       // 16-row tile index
    const int ncol0 = blockIdx.y * 64;             // first of 4 16-col tiles
    const int nlane = lane & 15;
    const int half  = lane >> 4;                   // 0 or 1
    const int koff  = half * 2;                    // K offset within 4-chunk

    // async-copy assignment: thread -> (row of 16, 16B column group of 16)
    const int prow = tid >> 4;                     // 0..15
    const int pcol = (tid & 15) * 4;               // 0..60 step 4

    const int mrow = mtile * 16 + nlane;           // A-row this lane loads
    const float* arow;
    if (GATHER) {
        arow = embed + (size_t)tokens[mrow] * (size_t)K;
    } else {
        arow = A + (size_t)mrow * (size_t)K;
    }

    v8f acc0 = {}, acc1 = {}, acc2 = {}, acc3 = {};

    const int nkb = K >> 4;                        // K / 16 (K is 512 or 640)

    // issue first tile fetch: 16 rows x 64 cols, one b128 per thread
    {
        const float*  gsrc = W + (size_t)prow * (size_t)N + ncol0 + pcol;
        unsigned ldst = (unsigned)(size_t)&wbuf[0][prow * WROW_ + pcol];
        asm volatile("global_load_async_to_lds_b128 %0, %1, off"
                     :: "v"(ldst), "v"(gsrc) : "memory");
    }

    for (int kb = 0; kb < nkb; ++kb) {
        async_wait0();          // my async stores to wbuf[kb&1] have landed
        __syncthreads();        // everyone's landed; prior reads of other buf drained

        if (kb + 1 < nkb) {
            const float*  gsrc =
                W + (size_t)((kb + 1) * 16 + prow) * (size_t)N + ncol0 + pcol;
            unsigned ldst = (unsigned)(size_t)&wbuf[(kb + 1) & 1][prow * WROW_ + pcol];
            asm volatile("global_load_async_to_lds_b128 %0, %1, off"
                         :: "v"(ldst), "v"(gsrc) : "memory");
        }

        const float* lb = &wbuf[kb & 1][0];
        const int kbase = kb * 16;
#pragma unroll
        for (int kk = 0; kk < 16; kk += 4) {
            v2f a;
            a.x = arow[kbase + kk + koff];
            a.y = arow[kbase + kk + koff + 1];
            const float* l = lb + (kk + koff) * WROW_ + nlane;
            v2f b0; b0.x = l[0];  b0.y = l[WROW_];
            v2f b1; b1.x = l[16]; b1.y = l[16 + WROW_];
            v2f b2; b2.x = l[32]; b2.y = l[32 + WROW_];
            v2f b3; b3.x = l[48]; b3.y = l[48 + WROW_];
            acc0 = wmma_f32_k4(a, b0, acc0);
            acc1 = wmma_f32_k4(a, b1, acc1);
            acc2 = wmma_f32_k4(a, b2, acc2);
            acc3 = wmma_f32_k4(a, b3, acc3);
        }
    }

    const float bb0 = bias[ncol0 + nlane];
    const float bb1 = bias[ncol0 + 16 + nlane];
    const float bb2 = bias[ncol0 + 32 + nlane];
    const float bb3 = bias[ncol0 + 48 + nlane];

#pragma unroll
    for (int r = 0; r < 8; ++r) {
        const int m = mtile * 16 + half * 8 + r;   // D layout: VGPR r -> M=r / M=r+8
        float* o = Out + (size_t)m * (size_t)N + ncol0 + nlane;
        o[0]  = acc0[r] + bb0;
        o[16] = acc1[r] + bb1;
        o[32] = acc2[r] + bb2;
        o[48] = acc3[r] + bb3;
    }
}

// ---------------------------------------------------------------------------
// One LSTM timestep.  grid = (B/16, H/16), block = 32 (one wave).
// Each wave owns one 16(batch) x 16(hidden) tile: computes all four gate
// tiles of  h_prev @ Wh  with f32 WMMA, adds precomputed Z = x@Wx + b,
// applies gates + mask, updates c in place, writes h_cur and Y[:,t,:].
// Wh (6.5 MB) stays resident in the 192 MB L2 across all 512 steps.
// ---------------------------------------------------------------------------
__global__ void lstm_step_kernel(const float* __restrict__ Zpre,   // (B*U, 4H)
                                 const float* __restrict__ Wh,     // (H, 4H)
                                 const float* __restrict__ h_prev, // (B, H)
                                 float* __restrict__ h_cur,        // (B, H)
                                 float* __restrict__ c,            // (B, H)
                                 float* __restrict__ Y,            // (B, U, H)
                                 const int* __restrict__ pred_len, // (B)
                                 int t) {
    const int lane  = threadIdx.x & 31;
    const int mtile = blockIdx.x;                  // batch tile (0..3)
    const int htile = blockIdx.y;                  // hidden tile (0..39)
    const int nlane = lane & 15;
    const int half  = lane >> 4;
    const int koff  = half * 2;

    const int arow = mtile * 16 + nlane;           // batch row for A frag
    const int ncol = htile * 16 + nlane;           // hidden col for B frags
    const float* aptr = h_prev + (size_t)arow * H_;

    v8f ai = {}, af = {}, ag = {}, ao = {};

    for (int k = 0; k < H_; k += 4) {
        v2f a;
        a.x = aptr[k + koff];
        a.y = aptr[k + koff + 1];
        const float* w = Wh + (size_t)(k + koff) * FH_ + ncol;
        v2f bi; bi.x = w[0];      bi.y = w[FH_];
        v2f bf; bf.x = w[H_];     bf.y = w[FH_ + H_];
        v2f bg; bg.x = w[2 * H_]; bg.y = w[FH_ + 2 * H_];
        v2f bo; bo.x = w[3 * H_]; bo.y = w[FH_ + 3 * H_];
        ai = wmma_f32_k4(a, bi, ai);
        af = wmma_f32_k4(a, bf, af);
        ag = wmma_f32_k4(a, bg, ag);
        ao = wmma_f32_k4(a, bo, ao);
    }

#pragma unroll
    for (int r = 0; r < 8; ++r) {
        const int b    = mtile * 16 + half * 8 + r;
        const int hcol = htile * 16 + nlane;
        const size_t zrow = (size_t)b * U_ + t;
        const float zi = ai[r] + Zpre[zrow * FH_ + hcol];
        const float zf = af[r] + Zpre[zrow * FH_ + H_ + hcol];
        const float zg = ag[r] + Zpre[zrow * FH_ + 2 * H_ + hcol];
        const float zo = ao[r] + Zpre[zrow * FH_ + 3 * H_ + hcol];

        const float ig = 1.0f / (1.0f + __expf(-zi));
        const float fg = 1.0f / (1.0f + __expf(-zf));
        const float og = 1.0f / (1.0f + __expf(-zo));
        const float gg = tanhf(zg);

        const size_t idx = (size_t)b * H_ + hcol;
        const float cold = c[idx];
        const float cnew = fg * cold + ig * gg;
        const float hnew = og * tanhf(cnew);

        const bool  msk = t < pred_len[b];
        const float hp  = h_prev[idx];
        const float h2  = msk ? hnew : hp;
        const float c2  = msk ? cnew : cold;

        c[idx]     = c2;
        h_cur[idx] = h2;
        Y[((size_t)b * U_ + t) * H_ + hcol] = h2;
    }
}

// ---------------------------------------------------------------------------
// In-place row layernorm over H_ elements.  grid = M_, block = 256.
// ---------------------------------------------------------------------------
__global__ void layernorm_kernel(float* __restrict__ Y,
                                 const float* __restrict__ g,
                                 const float* __restrict__ be) {
    const int row  = blockIdx.x;
    const int lane = threadIdx.x & 31;
    const int wave = threadIdx.x >> 5;
    float* y = Y + (size_t)row * H_;

    float s = 0.0f, s2 = 0.0f;
    for (int i = threadIdx.x; i < H_; i += 256) {
        const float v = y[i];
        s += v; s2 += v * v;
    }
#pragma unroll
    for (int off = 16; off > 0; off >>= 1) {
        s  += __shfl_down(s, off);
        s2 += __shfl_down(s2, off);
    }
    __shared__ float ss[8], sq[8];
    if (lane == 0) { ss[wave] = s; sq[wave] = s2; }
    __syncthreads();
    if (threadIdx.x == 0) {
        float S = 0.0f, S2 = 0.0f;
        for (int i = 0; i < 8; ++i) { S += ss[i]; S2 += sq[i]; }
        const float mean = S / (float)H_;
        float var = S2 / (float)H_ - mean * mean;
        var = fmaxf(var, 0.0f);
        ss[0] = mean;
        sq[0] = rsqrtf(var + LN_EPS);
    }
    __syncthreads();
    const float mean = ss[0];
    const float rstd = sq[0];
    for (int i = threadIdx.x; i < H_; i += 256) {
        y[i] = (y[i] - mean) * rstd * g[i] + be[i];
    }
}

__global__ void zero_kernel(float* __restrict__ p, int n) {
    const int i = blockIdx.x * blockDim.x + threadIdx.x;
    if (i < n) p[i] = 0.0f;
}

// ---------------------------------------------------------------------------
extern "C" void kernel_launch(void* const* d_in, const int* in_sizes, int n_in,
                              void* d_out, int out_size, void* d_ws, size_t ws_size,
                              hipStream_t stream) {
    (void)in_sizes; (void)n_in; (void)out_size; (void)ws_size;

    const int*   tokens   = (const int*)  d_in[0];
    const int*   pred_len = (const int*)  d_in[1];
    const float* embed    = (const float*)d_in[2];
    const float* W0x      = (const float*)d_in[3];
    const float* W0h      = (const float*)d_in[4];
    const float* b0       = (const float*)d_in[5];
    const float* g0       = (const float*)d_in[6];
    const float* be0      = (const float*)d_in[7];
    const float* P0w      = (const float*)d_in[8];
    const float* P0b      = (const float*)d_in[9];
    const float* W1x      = (const float*)d_in[10];
    const float* W1h      = (const float*)d_in[11];
    const float* b1       = (const float*)d_in[12];
    const float* g1       = (const float*)d_in[13];
    const float* be1      = (const float*)d_in[14];
    const float* P1w      = (const float*)d_in[15];
    const float* P1b      = (const float*)d_in[16];
    float* out = (float*)d_out;

    // Workspace layout (floats)
    float* ws = (float*)d_ws;
    const size_t ZN = (size_t)M_ * FH_;   // 83,886,080
    const size_t YN = (size_t)M_ * H_;    // 20,971,520
    const size_t XN = (size_t)M_ * P_;    // 16,777,216
    const size_t BH = (size_t)B_ * H_;    // 40,960
    float* Z  = ws;            // reused by both layers
    float* Y  = Z + ZN;        // reused by both layers
    float* X1 = Y + YN;
    float* h0 = X1 + XN;
    float* h1 = h0 + BH;
    float* cS = h1 + BH;

    const dim3 blk256(256);
    const dim3 gridZ(M_ / 128, FH_ / 64);   // (256, 40)
    const dim3 gridP(M_ / 128, P_ / 64);    // (256, 8)
    const dim3 gridS(B_ / 16, H_ / 16);     // (4, 40)
    const dim3 blkWave(32);
    const int  nState = (int)(3 * BH);      // h0,h1,c contiguous

    // ---------------- Layer 0 ----------------
    gemm_bias_kernel<true><<<gridZ, blk256, 0, stream>>>(
        nullptr, tokens, embed, W0x, b0, Z, M_, D_, FH_);
    zero_kernel<<<(nState + 255) / 256, blk256, 0, stream>>>(h0, nState);
    for (int t = 0; t < U_; ++t) {
        float* hp = (t & 1) ? h1 : h0;
        float* hc = (t & 1) ? h0 : h1;
        lstm_step_kernel<<<gridS, blkWave, 0, stream>>>(
            Z, W0h, hp, hc, cS, Y, pred_len, t);
    }
    layernorm_kernel<<<M_, blk256, 0, stream>>>(Y, g0, be0);
    gemm_bias_kernel<false><<<gridP, blk256, 0, stream>>>(
        Y, nullptr, nullptr, P0w, P0b, X1, M_, H_, P_);

    // ---------------- Layer 1 ----------------
    gemm_bias_kernel<false><<<gridZ, blk256, 0, stream>>>(
        X1, nullptr, nullptr, W1x, b1, Z, M_, P_, FH_);
    zero_kernel<<<(nState + 255) / 256, blk256, 0, stream>>>(h0, nState);
    for (int t = 0; t < U_; ++t) {
        float* hp = (t & 1) ? h1 : h0;
        float* hc = (t & 1) ? h0 : h1;
        lstm_step_kernel<<<gridS, blkWave, 0, stream>>>(
            Z, W1h, hp, hc, cS, Y, pred_len, t);
    }
    layernorm_kernel<<<M_, blk256, 0, stream>>>(Y, g1, be1);
    gemm_bias_kernel<false><<<gridP, blk256, 0, stream>>>(
        Y, nullptr, nullptr, P1w, P1b, out, M_, H_, P_);
}